// NL_model_88261577933600
// MI455X (gfx1250) — compile-verified
//
#include <hip/hip_runtime.h>
#include <hip/hip_bf16.h>

typedef __attribute__((ext_vector_type(2))) float v2f;
typedef __attribute__((ext_vector_type(8))) float v8f;

namespace {
constexpr int kN   = 50000;
constexpr int kE   = 1600000;
constexpr int kG   = 64;
constexpr int kD   = 32;   // hidden dim
constexpr int kDA  = 16;   // attr dim
constexpr int kRBF = 8;
}

__device__ __forceinline__ float siluf(float v) {
  return v / (1.0f + __expf(-v));
}

// ---------------- Kernel 1: node init  h = W_elem[x] @ W0 + b0 ----------------
__global__ __launch_bounds__(256) void nl_init_kernel(
    const int* __restrict__ x, const float* __restrict__ W_elem,
    const float* __restrict__ W0, const float* __restrict__ b0,
    float* __restrict__ h, float* __restrict__ xattr) {
  int idx = blockIdx.x * 256 + threadIdx.x;
  if (idx >= kN * kD) return;
  int n = idx >> 5;
  int d = idx & 31;
  int sp = x[n];
  const float* wrow = W_elem + sp * kDA;
  float acc = b0[d];
#pragma unroll
  for (int k = 0; k < kDA; ++k) acc += wrow[k] * W0[k * kD + d];
  h[idx] = acc;
  if (d < kDA) xattr[n * kDA + d] = wrow[d];
}

// ---------------- Kernel 2: edge message + scatter (WMMA f32 16x16x4) ----------------
// One wave = 16 edges. 32 v_wmma_f32_16x16x4_f32 per tile:
//   rw  = relu(rbf @ Wr1 + br1) @ Wr2   (first stage done in A-fragment layout via VALU)
//   msg = (h[src] @ Wmsg) * rw * sh_w   -> atomicAdd into agg[dst]
__global__ __launch_bounds__(256) void nl_edge_kernel(
    const int* __restrict__ edge_index, const float* __restrict__ pos,
    const float* __restrict__ period_vec,
    const float* __restrict__ Wr1, const float* __restrict__ br1,
    const float* __restrict__ Wr2, const float* __restrict__ Wmsg,
    const float* __restrict__ wsh,
    const float* __restrict__ h_in, float* __restrict__ agg, int layer) {
  __shared__ float sWr1[kRBF * kD];
  __shared__ float sbr1[kD];
  __shared__ float sWr2[kD * kD];
  __shared__ float sWmsg[kD * kD];

  int t = threadIdx.x;
  for (int k = t; k < kRBF * kD; k += 256) sWr1[k] = Wr1[layer * kRBF * kD + k];
  if (t < kD) sbr1[t] = br1[layer * kD + t];
  for (int k = t; k < kD * kD; k += 256) {
    sWr2[k]  = Wr2[layer * kD * kD + k];
    sWmsg[k] = Wmsg[layer * kD * kD + k];
  }
  __syncthreads();

  const float w0 = wsh[layer * 4 + 0], w1 = wsh[layer * 4 + 1];
  const float w2 = wsh[layer * 4 + 2], w3 = wsh[layer * 4 + 3];

  const int wave = t >> 5;
  const int lane = t & 31;
  const int lm   = lane & 15;   // row M within tile
  const int half = lane >> 4;   // lane-half (K offset for A/B fragments)
  const int tile = blockIdx.x * 8 + wave;
  if (tile * 16 >= kE) return;  // wave-uniform: EXEC stays all-1s for WMMA

  int  e     = tile * 16 + lm;
  bool valid = (e < kE);
  int  ec    = valid ? e : (kE - 1);

  // linear streams: prefetch ahead (global_prefetch_b8)
  __builtin_prefetch(edge_index + ec + 4096, 0, 1);
  __builtin_prefetch(edge_index + kE + ec + 4096, 0, 1);

  int src = edge_index[ec];
  int dst = edge_index[kE + ec];
  float dx = pos[dst * 3 + 0] - pos[src * 3 + 0] + period_vec[ec * 3 + 0];
  float dy = pos[dst * 3 + 1] - pos[src * 3 + 1] + period_vec[ec * 3 + 1];
  float dz = pos[dst * 3 + 2] - pos[src * 3 + 2] + period_vec[ec * 3 + 2];
  float len = sqrtf(dx * dx + dy * dy + dz * dz);
  float inv = 1.0f / (len + 1e-9f);
  float shw = w0 + (w1 * dx + w2 * dy + w3 * dz) * inv;  // edge_sh @ wsh[i]
  if (!valid) shw = 0.0f;  // dead lanes contribute exact zero

  // radial basis: centers = linspace(0,5,8), gamma = (8/5)^2
  float rbf[kRBF];
  const float cstep = 5.0f / 7.0f;
  const float gamma = (8.0f / 5.0f) * (8.0f / 5.0f);
#pragma unroll
  for (int k = 0; k < kRBF; ++k) {
    float dd = len - (float)k * cstep;
    rbf[k] = __expf(-gamma * dd * dd);
  }

  // tA = relu(rbf @ Wr1 + br1), produced directly in WMMA A-fragment layout:
  // step s, reg j -> column 4*s + j + 2*half of row lm
  float tA[16];
#pragma unroll
  for (int s = 0; s < 8; ++s) {
#pragma unroll
    for (int j = 0; j < 2; ++j) {
      int col = 4 * s + j + 2 * half;
      float acc = sbr1[col];
#pragma unroll
      for (int k = 0; k < kRBF; ++k) acc += rbf[k] * sWr1[k * kD + col];
      tA[2 * s + j] = acc > 0.0f ? acc : 0.0f;
    }
  }

  // gather h[src] row fragments (each row read exactly once across the 2 lane-halves)
  v2f hAf[8];
  const float* hrow = h_in + (size_t)src * kD + 2 * half;
#pragma unroll
  for (int s = 0; s < 8; ++s) {
    hAf[s].x = hrow[4 * s + 0];
    hAf[s].y = hrow[4 * s + 1];
  }

  v8f rw0 = {}, rw1 = {}, ms0 = {}, ms1 = {};
#pragma unroll
  for (int s = 0; s < 8; ++s) {
    int k0 = 4 * s + 2 * half;  // B fragment: K = k0 + j, N = lm (+16 for tile 1)
    v2f a;  a.x = tA[2 * s];  a.y = tA[2 * s + 1];
    v2f b0v; b0v.x = sWr2[k0 * kD + lm];       b0v.y = sWr2[(k0 + 1) * kD + lm];
    v2f b1v; b1v.x = sWr2[k0 * kD + 16 + lm];  b1v.y = sWr2[(k0 + 1) * kD + 16 + lm];
    rw0 = __builtin_amdgcn_wmma_f32_16x16x4_f32(false, a, false, b0v, (short)0, rw0, false, false);
    rw1 = __builtin_amdgcn_wmma_f32_16x16x4_f32(false, a, false, b1v, (short)0, rw1, false, false);
    v2f c0v; c0v.x = sWmsg[k0 * kD + lm];      c0v.y = sWmsg[(k0 + 1) * kD + lm];
    v2f c1v; c1v.x = sWmsg[k0 * kD + 16 + lm]; c1v.y = sWmsg[(k0 + 1) * kD + 16 + lm];
    ms0 = __builtin_amdgcn_wmma_f32_16x16x4_f32(false, hAf[s], false, c0v, (short)0, ms0, false, false);
    ms1 = __builtin_amdgcn_wmma_f32_16x16x4_f32(false, hAf[s], false, c1v, (short)0, ms1, false, false);
  }

  // epilogue: msg = (h_src@Wmsg) * rw * sh_w ; scatter into agg[dst]
#pragma unroll
  for (int r = 0; r < 8; ++r) {
    int   mrow = r + 8 * half;            // C-fragment row M for this VGPR
    float shwR = __shfl(shw, mrow, 32);
    int   dstR = __shfl(dst, mrow, 32);
    float m0 = ms0[r] * rw0[r] * shwR;
    float m1 = ms1[r] * rw1[r] * shwR;
    float* p = agg + (size_t)dstR * kD;
    __hip_atomic_fetch_add(p + lm,      m0, __ATOMIC_RELAXED, __HIP_MEMORY_SCOPE_AGENT);
    __hip_atomic_fetch_add(p + 16 + lm, m1, __ATOMIC_RELAXED, __HIP_MEMORY_SCOPE_AGENT);
  }
}

// ---------------- Kernel 3: node update  h = silu(agg + xattr@Wattr + h@Wself + b) ----------------
__global__ __launch_bounds__(256) void nl_node_kernel(
    const float* __restrict__ h_in, const float* __restrict__ agg,
    const float* __restrict__ xattr,
    const float* __restrict__ Wself, const float* __restrict__ Wattr,
    const float* __restrict__ bconv,
    float* __restrict__ h_out, int layer) {
  __shared__ float sWs[kD * kD];
  __shared__ float sWa[kDA * kD];
  __shared__ float sb[kD];
  int t = threadIdx.x;
  for (int k = t; k < kD * kD; k += 256) sWs[k] = Wself[layer * kD * kD + k];
  for (int k = t; k < kDA * kD; k += 256) sWa[k] = Wattr[layer * kDA * kD + k];
  if (t < kD) sb[t] = bconv[layer * kD + t];
  __syncthreads();
  int idx = blockIdx.x * 256 + t;
  if (idx >= kN * kD) return;
  int n = idx >> 5, d = idx & 31;
  float acc = agg[idx] + sb[d];
  const float* hr = h_in + (size_t)n * kD;
#pragma unroll
  for (int k = 0; k < kD; ++k) acc += hr[k] * sWs[k * kD + d];
  const float* xr = xattr + (size_t)n * kDA;
#pragma unroll
  for (int k = 0; k < kDA; ++k) acc += xr[k] * sWa[k * kD + d];
  h_out[idx] = siluf(acc);
}

// ---------------- Kernel 4: readout MLP + graph segment-sum ----------------
__global__ __launch_bounds__(256) void nl_head_kernel(
    const float* __restrict__ h, const int* __restrict__ batch,
    const float* __restrict__ Wp1, const float* __restrict__ bp1,
    const float* __restrict__ Wp2, const float* __restrict__ bp2,
    float* __restrict__ out) {
  __shared__ float sW1[kD * 16];
  __shared__ float sb1[16];
  __shared__ float sW2[16];
  __shared__ float sb2;
  int t = threadIdx.x;
  for (int k = t; k < kD * 16; k += 256) sW1[k] = Wp1[k];
  if (t < 16) { sb1[t] = bp1[t]; sW2[t] = Wp2[t]; }
  if (t == 0) sb2 = bp2[0];
  __syncthreads();
  int n = blockIdx.x * 256 + t;
  if (n >= kN) return;
  float hv[kD];
#pragma unroll
  for (int k = 0; k < kD; ++k) hv[k] = h[(size_t)n * kD + k];
  float acc2 = sb2;
#pragma unroll
  for (int j = 0; j < 16; ++j) {
    float a = sb1[j];
#pragma unroll
    for (int k = 0; k < kD; ++k) a += hv[k] * sW1[k * 16 + j];
    acc2 += siluf(a) * sW2[j];
  }
  // SCALE=1, SHIFT=0 -> identity on the graph sums
  __hip_atomic_fetch_add(out + batch[n], acc2, __ATOMIC_RELAXED, __HIP_MEMORY_SCOPE_AGENT);
}

// ---------------- host launcher ----------------
extern "C" void kernel_launch(void* const* d_in, const int* in_sizes, int n_in,
                              void* d_out, int out_size, void* d_ws, size_t ws_size,
                              hipStream_t stream) {
  (void)in_sizes; (void)n_in; (void)out_size; (void)ws_size;
  const int*   x          = (const int*)d_in[0];
  const float* pos        = (const float*)d_in[1];
  const int*   edge_index = (const int*)d_in[2];
  const float* period_vec = (const float*)d_in[3];
  const int*   batch      = (const int*)d_in[4];
  const float* W_elem = (const float*)d_in[5];
  const float* W0     = (const float*)d_in[6];
  const float* b0     = (const float*)d_in[7];
  const float* Wr1    = (const float*)d_in[8];
  const float* br1    = (const float*)d_in[9];
  const float* Wr2    = (const float*)d_in[10];
  const float* Wmsg   = (const float*)d_in[11];
  const float* Wattr  = (const float*)d_in[12];
  const float* Wself  = (const float*)d_in[13];
  const float* bconv  = (const float*)d_in[14];
  const float* wsh    = (const float*)d_in[15];
  const float* Wp1    = (const float*)d_in[16];
  const float* bp1    = (const float*)d_in[17];
  const float* Wp2    = (const float*)d_in[18];
  const float* bp2    = (const float*)d_in[19];

  float* hA    = (float*)d_ws;
  float* hB    = hA  + (size_t)kN * kD;
  float* agg   = hB  + (size_t)kN * kD;
  float* xattr = agg + (size_t)kN * kD;

  hipMemsetAsync(d_out, 0, kG * sizeof(float), stream);

  int nThreads = kN * kD;
  nl_init_kernel<<<(nThreads + 255) / 256, 256, 0, stream>>>(x, W_elem, W0, b0, hA, xattr);

  const float* hin = hA;
  float* hout = hB;
  int edgeBlocks = (kE / 16 + 7) / 8;  // 1 wave per 16-edge tile, 8 waves per block
  for (int i = 0; i < 3; ++i) {
    hipMemsetAsync(agg, 0, (size_t)kN * kD * sizeof(float), stream);
    nl_edge_kernel<<<edgeBlocks, 256, 0, stream>>>(edge_index, pos, period_vec,
                                                   Wr1, br1, Wr2, Wmsg, wsh, hin, agg, i);
    nl_node_kernel<<<(nThreads + 255) / 256, 256, 0, stream>>>(hin, agg, xattr,
                                                               Wself, Wattr, bconv, hout, i);
    const float* tmp = hin; hin = hout; hout = (float*)tmp;
  }
  nl_head_kernel<<<(kN + 255) / 256, 256, 0, stream>>>(hin, batch, Wp1, bp1, Wp2, bp2,
                                                       (float*)d_out);
}